// TTTRNNCell_46909632807528
// MI455X (gfx1250) — compile-verified
//
#include <hip/hip_runtime.h>
#include <hip/hip_bf16.h>
#include <stdint.h>

// ---------------------------------------------------------------------------
// TTT-RNN cell on MI455X (gfx1250): full f32 pipeline on V_WMMA_F32_16X16X4_F32
// with GLOBAL_LOAD_ASYNC_TO_LDS_B128 double-buffered tile staging (ASYNCcnt).
// Workload is memory-bound (~150 GF vs ~1.2 GB HBM traffic @ 23.3 TB/s), so we
// keep full f32 precision and spend effort on the data-movement path:
// async global->LDS copies overlap the next tile's HBM latency with the
// current tile's WMMAs, with a single workgroup barrier per k-stage.
// ---------------------------------------------------------------------------

typedef __attribute__((ext_vector_type(2))) float v2f;
typedef __attribute__((ext_vector_type(8))) float v8f;

#define B_DIM 32768
#define F_DIM 256
#define U_DIM 256
#define H_DIM 1024   // 4*U
#define ZK    512    // F+U
#define LR_C  0.01f
#define EPS_C 1e-3f

__device__ __forceinline__ v8f wmma4(v2f a, v2f b, v8f c) {
  // V_WMMA_F32_16X16X4_F32: (neg_a, A, neg_b, B, c_mod, C, reuse_a, reuse_b)
  return __builtin_amdgcn_wmma_f32_16x16x4_f32(false, a, false, b, (short)0, c,
                                               false, false);
}

// Async global->LDS copy of 16 bytes per lane (tracked by ASYNCcnt).
// lds is the low 32 bits of a generic pointer into __shared__ (the LDS
// aperture keeps the wave-relative LDS byte address in addr[31:0]).
__device__ __forceinline__ void async_b128(uint32_t lds, const float* gaddr) {
  asm volatile("global_load_async_to_lds_b128 %0, %1, off"
               :: "v"(lds), "v"(gaddr)
               : "memory");
}
__device__ __forceinline__ void wait_async0() {
  asm volatile("s_wait_asynccnt 0x0" ::: "memory");
}
__device__ __forceinline__ uint32_t lds_addr32(const void* p) {
  return (uint32_t)(uintptr_t)p;
}

__device__ __forceinline__ float gelu_f(float x) {
  return 0.5f * x * (1.0f + erff(x * 0.70710678118654752f));
}
__device__ __forceinline__ float gelu_grad_f(float x) {
  float cdf = 0.5f * (1.0f + erff(x * 0.70710678118654752f));
  return cdf + x * 0.3989422804014327f * __expf(-0.5f * x * x);
}

enum {
  EPI_NONE = 0,
  EPI_BIAS = 1,
  EPI_BIAS_GELU = 2,
  EPI_BIAS_GELU_PRE = 3,  // C = acc+bias, C2 = gelu(C)
  EPI_BIAS_AUX = 4,       // C = acc+bias+auxScale*aux
  EPI_GELUGRAD = 5        // C = acc * gelu'(aux)
};

// ---------------------------------------------------------------------------
// Generic tiled WMMA GEMM:  C[M,N] = epilogue(A[M,K] @ B + bias)
// TRANSB=false: B is [K,N] row-major.  TRANSB=true: B is [N,K] row-major,
// used as B^T (element (k,n) = Bst[n*K+k]).
// Block tile 128x128, BK=16, 256 threads (8 waves, 4(M) x 2(N)); each wave
// owns a 32x64 tile = 8 f32 accumulators -> 32 v_wmma per k-stage.
// LDS layouts:
//   As: [2][128][20] m-major. 80B row: 16B-aligned for async b128 staging,
//       8B-aligned + bank-conflict-free for single ds_load_b64 A-fragments.
//   Bs (TRANSB):  [2][128*20] n-major, same properties, b64 B-fragments.
//   Bs (!TRANSB): [2][16*132] k-major, 528B row (16B-aligned staging);
//       (k,k+1) fragment pair merges into one ds_load_2addr_b32.
// ---------------------------------------------------------------------------
template <int EPI, bool TRANSB>
__global__ __launch_bounds__(256) void gemm_wmma(
    const float* __restrict__ A, const float* __restrict__ Bm,
    const float* __restrict__ bias, const float* __restrict__ aux,
    float* __restrict__ C, float* __restrict__ C2, int M, int N, int K,
    float auxScale) {
  __shared__ float As[2][128][20];
  __shared__ float Bsh[2][TRANSB ? (128 * 20) : (16 * 132)];

  const int tid = threadIdx.x;
  const int lane = tid & 31;
  const int wave = tid >> 5;
  const int waveM = wave & 3;   // 4 waves along M
  const int waveN = wave >> 2;  // 2 waves along N
  const int blockM = blockIdx.y * 128;
  const int blockN = blockIdx.x * 128;
  const int lm = lane & 15;         // row/col within a 16-wide tile
  const int kk = (lane >> 4) << 1;  // upper half-wave handles K+2,K+3

  v8f acc[2][4] = {};

  // staging coordinates
  const int ai = tid >> 1;        // A tile row 0..127
  const int aj = (tid & 1) << 3;  // 8 contiguous K floats (two b128)
  const float* Abase = A + (size_t)(blockM + ai) * K + aj;
  const int bn = tid >> 1;         // TRANSB: stored-B row 0..127
  const int bk = (tid & 1) << 3;
  const int bi = tid >> 4;         // !TRANSB: k row 0..15
  const int bj = (tid & 15) << 3;  // 8 contiguous N floats
  const float* Bbase = TRANSB ? (Bm + (size_t)(blockN + bn) * K + bk)
                              : (Bm + (size_t)bi * N + blockN + bj);

  auto stage = [&](int buf, int k0) {
    const uint32_t la = lds_addr32(&As[buf][ai][aj]);
    const float* ga = Abase + k0;
    async_b128(la, ga);
    async_b128(la + 16u, ga + 4);
    uint32_t lb;
    const float* gb;
    if (TRANSB) {
      lb = lds_addr32(&Bsh[buf][bn * 20 + bk]);
      gb = Bbase + k0;
    } else {
      lb = lds_addr32(&Bsh[buf][bi * 132 + bj]);
      gb = Bbase + (size_t)k0 * N;
    }
    async_b128(lb, gb);
    async_b128(lb + 16u, gb + 4);
  };

  stage(0, 0);
  int buf = 0;
  for (int k0 = 0; k0 < K; k0 += 16, buf ^= 1) {
    wait_async0();        // own async copies into `buf` complete
    __syncthreads();      // everyone's copies visible; prev compute retired
    if (k0 + 16 < K) stage(buf ^ 1, k0 + 16);  // prefetch next stage

#pragma unroll
    for (int ks = 0; ks < 16; ks += 4) {
      v2f af[2], bf[4];
#pragma unroll
      for (int im = 0; im < 2; ++im) {
        const int m = waveM * 32 + im * 16 + lm;
        af[im] = *(const v2f*)&As[buf][m][ks + kk];
      }
#pragma unroll
      for (int in = 0; in < 4; ++in) {
        const int n = waveN * 64 + in * 16 + lm;
        if (TRANSB) {
          bf[in] = *(const v2f*)&Bsh[buf][n * 20 + ks + kk];
        } else {
          bf[in][0] = Bsh[buf][(ks + kk) * 132 + n];
          bf[in][1] = Bsh[buf][(ks + kk + 1) * 132 + n];
        }
      }
#pragma unroll
      for (int im = 0; im < 2; ++im)
#pragma unroll
        for (int in = 0; in < 4; ++in)
          acc[im][in] = wmma4(af[im], bf[in], acc[im][in]);
    }
  }

  // --- epilogue: C/D layout: VGPR i -> M = i + 8*(lane>=16), N = lane&15 ---
  const int mh = (lane >> 4) << 3;
#pragma unroll
  for (int im = 0; im < 2; ++im) {
#pragma unroll
    for (int in = 0; in < 4; ++in) {
      const int n = blockN + waveN * 64 + in * 16 + lm;
      const float bv = (EPI == EPI_NONE || EPI == EPI_GELUGRAD) ? 0.f : bias[n];
#pragma unroll
      for (int i = 0; i < 8; ++i) {
        const int m = blockM + waveM * 32 + im * 16 + mh + i;
        const size_t off = (size_t)m * N + n;
        const float v = acc[im][in][i];
        if (EPI == EPI_NONE) {
          C[off] = v;
        } else if (EPI == EPI_BIAS) {
          C[off] = v + bv;
        } else if (EPI == EPI_BIAS_GELU) {
          C[off] = gelu_f(v + bv);
        } else if (EPI == EPI_BIAS_GELU_PRE) {
          const float pre = v + bv;
          C[off] = pre;
          C2[off] = gelu_f(pre);
        } else if (EPI == EPI_BIAS_AUX) {
          C[off] = v + bv + auxScale * aux[off];
        } else {  // EPI_GELUGRAD
          C[off] = v * gelu_grad_f(aux[off]);
        }
      }
    }
  }
}

// ---------------------------------------------------------------------------
// Weight-gradient GEMM, split-K over the batch dimension:
//   C[M,N] += scale * sum_b A[b,m] * Bm[b,n]     (A: [rows,M], Bm: [rows,N])
// Tile 64x64, 256 threads (8 waves: 4(M) x 2(N), wave tile 16x32).
// Async b128 double-buffered staging; partial results folded into C via
// global f32 atomics (SGD update folded in via `scale`).
// LDS tiles k-major [16][68]: 272B rows -> 16B-aligned staging, (k,k+1)
// fragment pairs merge into ds_load_2addr_b32, conflict-free banks.
// ---------------------------------------------------------------------------
__global__ __launch_bounds__(256) void gemm_atn_wmma(
    const float* __restrict__ A, const float* __restrict__ Bm,
    float* __restrict__ C, int M, int N, int chunkRows, float scale) {
  __shared__ float Ats[2][16][68];
  __shared__ float Bts[2][16][68];

  const int tid = threadIdx.x;
  const int lane = tid & 31;
  const int wave = tid >> 5;
  const int waveM = wave & 3;
  const int waveN = wave >> 2;
  const int m0 = blockIdx.y * 64;
  const int n0 = blockIdx.x * 64;
  const int r0 = blockIdx.z * chunkRows;
  const int lm = lane & 15;
  const int kk = (lane >> 4) << 1;

  v8f acc[2] = {};

  const int li = tid >> 4;         // batch row 0..15
  const int lj = (tid & 15) << 2;  // 4 contiguous columns (one b128)
  const float* Abase = A + (size_t)(r0 + li) * M + m0 + lj;
  const float* Bbase = Bm + (size_t)(r0 + li) * N + n0 + lj;

  auto stage = [&](int buf, int b) {
    async_b128(lds_addr32(&Ats[buf][li][lj]), Abase + (size_t)b * M);
    async_b128(lds_addr32(&Bts[buf][li][lj]), Bbase + (size_t)b * N);
  };

  stage(0, 0);
  int buf = 0;
  for (int b = 0; b < chunkRows; b += 16, buf ^= 1) {
    wait_async0();
    __syncthreads();
    if (b + 16 < chunkRows) stage(buf ^ 1, b + 16);

#pragma unroll
    for (int ks = 0; ks < 16; ks += 4) {
      v2f af, bf0, bf1;
      const int m = waveM * 16 + lm;
      af[0] = Ats[buf][ks + kk][m];
      af[1] = Ats[buf][ks + kk + 1][m];
      const int nt = waveN * 32 + lm;
      bf0[0] = Bts[buf][ks + kk][nt];
      bf0[1] = Bts[buf][ks + kk + 1][nt];
      bf1[0] = Bts[buf][ks + kk][nt + 16];
      bf1[1] = Bts[buf][ks + kk + 1][nt + 16];
      acc[0] = wmma4(af, bf0, acc[0]);
      acc[1] = wmma4(af, bf1, acc[1]);
    }
  }

  const int mh = (lane >> 4) << 3;
#pragma unroll
  for (int in = 0; in < 2; ++in) {
    const int n = n0 + waveN * 32 + in * 16 + lm;
#pragma unroll
    for (int i = 0; i < 8; ++i) {
      const int m = m0 + waveM * 16 + mh + i;
      atomicAdd(&C[(size_t)m * N + n], scale * acc[in][i]);
    }
  }
}

// ---------------------------------------------------------------------------
// Fused LayerNorm forward + backward (one wave32 per row, 8 rows per block).
//   recon = LN(rpre)*gamma+beta ; dy = 2*(recon-phi)/(B*U)
//   dx    = rstd*(dy*g - mean(dy*g) - xhat*mean(dy*g*xhat))
// ---------------------------------------------------------------------------
__device__ __forceinline__ float wave_sum(float v) {
#pragma unroll
  for (int o = 16; o > 0; o >>= 1) v += __shfl_xor(v, o, 32);
  return v;
}

__global__ __launch_bounds__(256) void ln_fwd_bwd(
    const float* __restrict__ rpre, const float* __restrict__ phi,
    const float* __restrict__ gamma, const float* __restrict__ beta,
    float* __restrict__ recon, float* __restrict__ dx) {
  const int lane = threadIdx.x & 31;
  const int wave = threadIdx.x >> 5;
  const int row = blockIdx.x * 8 + wave;
  const float* x = rpre + (size_t)row * U_DIM;
  const float* t = phi + (size_t)row * U_DIM;

  float v[8];
  float s = 0.f, s2 = 0.f;
#pragma unroll
  for (int j = 0; j < 8; ++j) {
    v[j] = x[lane + 32 * j];
    s += v[j];
    s2 += v[j] * v[j];
  }
  s = wave_sum(s);
  s2 = wave_sum(s2);
  const float mean = s * (1.f / 256.f);
  const float var = s2 * (1.f / 256.f) - mean * mean;
  const float rstd = rsqrtf(var + EPS_C);

  float xh[8], dxh[8];
  float c1 = 0.f, c2 = 0.f;
  const float dscale = 2.f / ((float)B_DIM * (float)U_DIM);
#pragma unroll
  for (int j = 0; j < 8; ++j) {
    const int c = lane + 32 * j;
    xh[j] = (v[j] - mean) * rstd;
    const float g = gamma[c];
    const float y = xh[j] * g + beta[c];
    recon[(size_t)row * U_DIM + c] = y;
    const float dy = dscale * (y - t[c]);
    dxh[j] = dy * g;
    c1 += dxh[j];
    c2 += dxh[j] * xh[j];
  }
  c1 = wave_sum(c1) * (1.f / 256.f);
  c2 = wave_sum(c2) * (1.f / 256.f);
#pragma unroll
  for (int j = 0; j < 8; ++j) {
    const int c = lane + 32 * j;
    dx[(size_t)row * U_DIM + c] = rstd * (dxh[j] - c1 - xh[j] * c2);
  }
}

// --------------------------- small glue kernels ----------------------------
__global__ void concat_z(const float* __restrict__ x,
                         const float* __restrict__ h, float* __restrict__ z) {
  const size_t i = (size_t)blockIdx.x * 256 + threadIdx.x;  // over B*256
  const size_t b = i >> 8;
  const int c = (int)(i & 255);
  z[b * ZK + c] = x[i];
  z[b * ZK + U_DIM + c] = h[i];
}

__global__ void copy_f32(const float* __restrict__ in, float* __restrict__ out,
                         int n) {
  const int i = blockIdx.x * 256 + threadIdx.x;
  if (i < n) out[i] = in[i];
}

// bias-gradient column sum, folded SGD update: bu[c] += scale * sum_r X[r,c]
__global__ void colsum_update(const float* __restrict__ X,
                              float* __restrict__ bu, int N, int rowsPerChunk,
                              float scale) {
  const int c = blockIdx.x * 256 + threadIdx.x;
  const int r0 = blockIdx.y * rowsPerChunk;
  float s = 0.f;
  for (int r = 0; r < rowsPerChunk; ++r) s += X[(size_t)(r0 + r) * N + c];
  atomicAdd(&bu[c], scale * s);
}

// ---------------------------------------------------------------------------
extern "C" void kernel_launch(void* const* d_in, const int* in_sizes, int n_in,
                              void* d_out, int out_size, void* d_ws,
                              size_t ws_size, hipStream_t stream) {
  (void)in_sizes; (void)n_in; (void)out_size; (void)ws_size;
  const float* x      = (const float*)d_in[0];
  const float* h_prev = (const float*)d_in[1];
  const float* W_phi  = (const float*)d_in[2];
  const float* b_phi  = (const float*)d_in[3];
  const float* W_psi  = (const float*)d_in[4];
  const float* b_psi  = (const float*)d_in[5];
  const float* W1     = (const float*)d_in[6];
  const float* b1     = (const float*)d_in[7];
  const float* W2     = (const float*)d_in[8];
  const float* b2     = (const float*)d_in[9];
  const float* W_g    = (const float*)d_in[10];
  const float* g_bias = (const float*)d_in[11];
  const float* gamma  = (const float*)d_in[12];
  const float* beta   = (const float*)d_in[13];
  const float* W_h    = (const float*)d_in[14];
  const float* h_bias = (const float*)d_in[15];
  float* out = (float*)d_out;

  float* p = (float*)d_ws;
  auto alloc = [&](size_t n) { float* r = p; p += n; return r; };
  float* z     = alloc((size_t)B_DIM * ZK);
  float* phi   = alloc((size_t)B_DIM * U_DIM);
  float* pre1  = alloc((size_t)B_DIM * H_DIM);
  float* a1    = alloc((size_t)B_DIM * H_DIM);  // reused: a1b after backward
  float* e2    = alloc((size_t)B_DIM * U_DIM);  // reused: feat
  float* rpre  = alloc((size_t)B_DIM * U_DIM);
  float* recon = alloc((size_t)B_DIM * U_DIM);
  float* drpre = alloc((size_t)B_DIM * U_DIM);  // reused: psi
  float* de2   = alloc((size_t)B_DIM * U_DIM);
  float* dpre1 = alloc((size_t)B_DIM * H_DIM);
  float* W1u   = alloc((size_t)U_DIM * H_DIM);
  float* W2u   = alloc((size_t)H_DIM * U_DIM);
  float* b1u   = alloc(H_DIM);
  float* b2u   = alloc(U_DIM);

  const dim3 blk(256);

  // z = concat(x, h_prev)
  concat_z<<<dim3(B_DIM), blk, 0, stream>>>(x, h_prev, z);
  // phi = z @ W_phi + b_phi
  gemm_wmma<EPI_BIAS, false><<<dim3(U_DIM / 128, B_DIM / 128), blk, 0, stream>>>(
      z, W_phi, b_phi, nullptr, phi, nullptr, B_DIM, U_DIM, ZK, 0.f);
  // pre1 = phi @ W1 + b1 ; a1 = gelu(pre1)
  gemm_wmma<EPI_BIAS_GELU_PRE, false>
      <<<dim3(H_DIM / 128, B_DIM / 128), blk, 0, stream>>>(
          phi, W1, b1, nullptr, pre1, a1, B_DIM, H_DIM, U_DIM, 0.f);
  // e2 = a1 @ W2 + b2
  gemm_wmma<EPI_BIAS, false><<<dim3(U_DIM / 128, B_DIM / 128), blk, 0, stream>>>(
      a1, W2, b2, nullptr, e2, nullptr, B_DIM, U_DIM, H_DIM, 0.f);
  // rpre = e2 @ W_g + g_bias
  gemm_wmma<EPI_BIAS, false><<<dim3(U_DIM / 128, B_DIM / 128), blk, 0, stream>>>(
      e2, W_g, g_bias, nullptr, rpre, nullptr, B_DIM, U_DIM, U_DIM, 0.f);
  // recon = LN(rpre); drpre = dL/d rpre
  ln_fwd_bwd<<<dim3(B_DIM / 8), blk, 0, stream>>>(rpre, phi, gamma, beta, recon,
                                                  drpre);
  // de2 = drpre @ W_g^T
  gemm_wmma<EPI_NONE, true><<<dim3(U_DIM / 128, B_DIM / 128), blk, 0, stream>>>(
      drpre, W_g, nullptr, nullptr, de2, nullptr, B_DIM, U_DIM, U_DIM, 0.f);
  // W2u = W2 - LR * a1^T @ de2 ; b2u = b2 - LR * colsum(de2)
  copy_f32<<<dim3(H_DIM * U_DIM / 256), blk, 0, stream>>>(W2, W2u,
                                                          H_DIM * U_DIM);
  copy_f32<<<dim3(1), blk, 0, stream>>>(b2, b2u, U_DIM);
  gemm_atn_wmma<<<dim3(U_DIM / 64, H_DIM / 64, 16), blk, 0, stream>>>(
      a1, de2, W2u, H_DIM, U_DIM, B_DIM / 16, -LR_C);
  colsum_update<<<dim3(U_DIM / 256, 32), blk, 0, stream>>>(de2, b2u, U_DIM,
                                                           B_DIM / 32, -LR_C);
  // dpre1 = (de2 @ W2^T) * gelu'(pre1)   (uses ORIGINAL W2)
  gemm_wmma<EPI_GELUGRAD, true>
      <<<dim3(H_DIM / 128, B_DIM / 128), blk, 0, stream>>>(
          de2, W2, nullptr, pre1, dpre1, nullptr, B_DIM, H_DIM, U_DIM, 0.f);
  // W1u = W1 - LR * phi^T @ dpre1 ; b1u = b1 - LR * colsum(dpre1)
  copy_f32<<<dim3(U_DIM * H_DIM / 256), blk, 0, stream>>>(W1, W1u,
                                                          U_DIM * H_DIM);
  copy_f32<<<dim3(H_DIM / 256), blk, 0, stream>>>(b1, b1u, H_DIM);
  gemm_atn_wmma<<<dim3(H_DIM / 64, U_DIM / 64, 16), blk, 0, stream>>>(
      phi, dpre1, W1u, U_DIM, H_DIM, B_DIM / 16, -LR_C);
  colsum_update<<<dim3(H_DIM / 256, 32), blk, 0, stream>>>(dpre1, b1u, H_DIM,
                                                           B_DIM / 32, -LR_C);
  // psi = z @ W_psi + b_psi   (reuse drpre buffer)
  float* psi = drpre;
  gemm_wmma<EPI_BIAS, false><<<dim3(U_DIM / 128, B_DIM / 128), blk, 0, stream>>>(
      z, W_psi, b_psi, nullptr, psi, nullptr, B_DIM, U_DIM, ZK, 0.f);
  // a1b = gelu(psi @ W1u + b1u)   (reuse a1 buffer)
  gemm_wmma<EPI_BIAS_GELU, false>
      <<<dim3(H_DIM / 128, B_DIM / 128), blk, 0, stream>>>(
          psi, W1u, b1u, nullptr, a1, nullptr, B_DIM, H_DIM, U_DIM, 0.f);
  // feat = a1b @ W2u + b2u + 0.1*recon   (reuse e2 buffer)
  gemm_wmma<EPI_BIAS_AUX, false>
      <<<dim3(U_DIM / 128, B_DIM / 128), blk, 0, stream>>>(
          a1, W2u, b2u, recon, e2, nullptr, B_DIM, U_DIM, H_DIM, 0.1f);
  // out = feat @ W_h + h_bias
  gemm_wmma<EPI_BIAS, false><<<dim3(U_DIM / 128, B_DIM / 128), blk, 0, stream>>>(
      e2, W_h, h_bias, nullptr, out, nullptr, B_DIM, U_DIM, U_DIM, 0.f);
}